// LeakyAvg_14534169329970
// MI455X (gfx1250) — compile-verified
//
#include <hip/hip_runtime.h>
#include <hip/hip_bf16.h>
#include <math.h>

// LeakyAvg: out[b,h,t,d] = sum_{s<=t} exp(-beta_h*(t-s)) * k[b,h,s,d]
// Implemented as a chunked (16-row) scan:
//   intra-chunk: 16x16 lower-triangular decay matrix L times k-chunk via
//                4x V_WMMA_F32_16X16X4_F32 (K accumulation)
//   inter-chunk: out[m][n] += exp(-beta*(m+1)) * carry[n],
//                carry = previous chunk's last output row (D VGPR7, hi lanes)
// T is split into NPART partitions; each (except part 0) warms up 256 rows,
// which is exact in fp32 because exp(-beta*256) underflows to 0 (beta >= 0.5).

typedef float v2f __attribute__((ext_vector_type(2)));
typedef float v8f __attribute__((ext_vector_type(8)));

#define T_DIM   2048
#define HS      64
#define NH      16
#define CH      16                  // chunk = one WMMA M-tile
#define NPART   4
#define PART_T  (T_DIM / NPART)     // 512 rows per partition
#define WARM_CH 16                  // 256-row warmup (exact: exp(-128) -> 0 in fp32)

__global__ __launch_bounds__(32)
void LeakyAvg_scan_wmma(const float* __restrict__ K,
                        const float* __restrict__ lkb,
                        float* __restrict__ Out) {
    const int lane = threadIdx.x;          // 0..31 (wave32)
    const int bh   = blockIdx.x;           // b*NH + h
    const int h    = bh & (NH - 1);
    const int dg   = blockIdx.y;           // d-group: 16 columns each
    const int part = blockIdx.z;           // T partition
    const int d0   = dg * 16;

    const float beta = fabsf(lkb[h]) * 10.0f;

    const int hi  = (lane >= 16) ? 1 : 0;  // which half-wave
    const int lid = lane & 15;

    // ---- constant A operand: decay[m][s] = (m>=s) ? exp(-beta*(m-s)) : 0
    // 16x4 f32 A layout: VGPR0 holds K={0|2}, VGPR1 holds K={1|3}; M = lane%16.
    v2f a[4];
#pragma unroll
    for (int kk = 0; kk < 4; ++kk) {
        const int s0 = 4 * kk + 2 * hi;
        const int m  = lid;
        a[kk].x = (m >= s0    ) ? __expf(-beta * (float)(m - s0    )) : 0.0f;
        a[kk].y = (m >= s0 + 1) ? __expf(-beta * (float)(m - s0 - 1)) : 0.0f;
    }

    // ---- carry factors per D VGPR: row m = i + 8*hi  ->  exp(-beta*(m+1))
    v8f fac;
#pragma unroll
    for (int i = 0; i < 8; ++i)
        fac[i] = __expf(-beta * (float)(i + 8 * hi + 1));

    const float* kbase = K   + (size_t)bh * T_DIM * HS;
    float*       obase = Out + (size_t)bh * T_DIM * HS;

    const int t_out0 = part * PART_T;
    const int warm   = (part == 0) ? 0 : WARM_CH;
    float carry = 0.0f;

    for (int ci = -warm; ci < PART_T / CH; ++ci) {
        const int t0 = t_out0 + ci * CH;

        // ---- load B operand: k tile rows [t0..t0+15], cols [d0..d0+15]
        // 4x16 f32 B layout per K-slice kk: VGPR0 = row {4kk|4kk+2},
        // VGPR1 = row {4kk+1|4kk+3}; N = lane%16 across lanes.
        const float* kt = kbase + (size_t)t0 * HS + d0 + lid;
        v2f b[4];
#pragma unroll
        for (int kk = 0; kk < 4; ++kk) {
            const int r = 4 * kk + 2 * hi;
            b[kk].x = kt[(size_t)r * HS];
            b[kk].y = kt[(size_t)(r + 1) * HS];
        }
        // prefetch next chunk's tile (global_prefetch_b8)
        __builtin_prefetch(kt + (size_t)CH * HS, 0, 1);

        // ---- intra-chunk: C = L16 x Ktile via 4 chained K=4 WMMAs
        v8f c = {};
#pragma unroll
        for (int kk = 0; kk < 4; ++kk) {
            c = __builtin_amdgcn_wmma_f32_16x16x4_f32(
                    /*neg_a=*/false, a[kk], /*neg_b=*/false, b[kk],
                    /*c_mod=*/(short)0, c, /*reuse_a=*/false, /*reuse_b=*/false);
        }

        // ---- inter-chunk carry: out[m][n] = c[m][n] + exp(-beta*(m+1))*carry[n]
#pragma unroll
        for (int i = 0; i < 8; ++i)
            c[i] += fac[i] * carry;

        // new carry = out row m=15 (D VGPR7, lanes 16..31); broadcast to both halves
        carry = __shfl(c[7], lid + 16, 32);

        // ---- store output tile (suppressed for warmup chunks)
        if (ci >= 0) {
            float* ot = obase + (size_t)t0 * HS + d0 + lid;
#pragma unroll
            for (int i = 0; i < 8; ++i)
                ot[(size_t)(i + 8 * hi) * HS] = c[i];
        }
    }
}

extern "C" void kernel_launch(void* const* d_in, const int* in_sizes, int n_in,
                              void* d_out, int out_size, void* d_ws, size_t ws_size,
                              hipStream_t stream) {
    const float* K   = (const float*)d_in[0];   // (B, NH, T, HS) fp32
    const float* lkb = (const float*)d_in[1];   // (1, NH, 1, 1) fp32
    float* out = (float*)d_out;

    const int B = in_sizes[0] / (NH * T_DIM * HS);
    dim3 grid(B * NH, HS / 16, NPART);
    LeakyAvg_scan_wmma<<<grid, 32, 0, stream>>>(K, lkb, out);
}